// MeshThickness_49581102465727
// MI455X (gfx1250) — compile-verified
//
#include <hip/hip_runtime.h>
#include <hip/hip_bf16.h>
#include <math.h>

// ---------------------------------------------------------------------------
// MeshThickness fused pipeline for gfx1250 (MI455X, wave32)
//  - V_WMMA_F32_16X16X4_F32 for the K=3 dot-product score tiles
//  - Tensor Data Mover (tensor_load_to_lds) double-buffers 128-face blocks
//    of packed {centroid, |c|^2, normal} records into LDS (TENSORcnt)
//  - raw v_sqrt_f32 for ranking-only sqrt (precise sqrtf kept for outputs)
// ---------------------------------------------------------------------------

#define RCOEF  0.2f
#define KSEL   10
#define TILE_F 128           // faces per TDM stage (32B/face -> 4KB per stage)

typedef float v2f __attribute__((ext_vector_type(2)));
typedef float v8f __attribute__((ext_vector_type(8)));
typedef unsigned int u32x4 __attribute__((ext_vector_type(4)));
typedef int i32x4 __attribute__((ext_vector_type(4)));
typedef int i32x8 __attribute__((ext_vector_type(8)));

__device__ __forceinline__ v8f wmma_f32_16x16x4(v2f a, v2f b, v8f c) {
  // 8 args: (neg_a, A, neg_b, B, c_mod, C, reuse_a, reuse_b)
  return __builtin_amdgcn_wmma_f32_16x16x4_f32(
      false, a, false, b, (short)0, c, false, false);
}

// TDM: 1-D contiguous tile of `nelem` dwords from global `ga` to LDS `lds_off`.
// Descriptor layout per CDNA5 ISA ch.8 (count=1, type=2, data_size=4B,
// workgroup_mask=0 [not in cluster], no pad/iterate/gather).
// This toolchain exposes the 6-arg builtin: (v4u, v8i, v4i, v4i, v8i, cpol).
__device__ __forceinline__ void tdm_load_1d(unsigned long long ga,
                                            unsigned lds_off,
                                            unsigned nelem,
                                            unsigned total_elems) {
  u32x4 g0;
  g0[0] = 1u;                                              // count=1
  g0[1] = lds_off;                                         // lds_addr
  g0[2] = (unsigned)ga;                                    // global_addr[31:0]
  g0[3] = (unsigned)((ga >> 32) & 0x01FFFFFFull)           // global_addr[56:32]
        | (2u << 30);                                      // type=2 (image)
  i32x8 g1;
  g1[0] = (int)(2u << 16);                                 // data_size=4B, mask=0
  g1[1] = (int)((total_elems & 0xFFFFu) << 16);            // tensor_dim0[15:0]
  g1[2] = (int)(((total_elems >> 16) & 0xFFFFu)            // tensor_dim0[31:16]
        | (1u << 16));                                     // tensor_dim1=1
  g1[3] = (int)((nelem & 0xFFFFu) << 16);                  // tile_dim0
  g1[4] = 0;                                               // tile_dim1/2 unused
  g1[5] = (int)total_elems;                                // dim0_stride[31:0]
  g1[6] = 0;
  g1[7] = 0;
  i32x4 gz4; gz4[0] = 0; gz4[1] = 0; gz4[2] = 0; gz4[3] = 0;
  i32x8 gz8;
  gz8[0] = 0; gz8[1] = 0; gz8[2] = 0; gz8[3] = 0;
  gz8[4] = 0; gz8[5] = 0; gz8[6] = 0; gz8[7] = 0;
  __builtin_amdgcn_tensor_load_to_lds(g0, g1, gz4, gz4, gz8, 0);
}

struct F3 { float x, y, z; };
__device__ __forceinline__ F3 f3sub(F3 a, F3 b) { F3 r{a.x - b.x, a.y - b.y, a.z - b.z}; return r; }
__device__ __forceinline__ float f3dot(F3 a, F3 b) { return a.x * b.x + a.y * b.y + a.z * b.z; }
__device__ __forceinline__ F3 f3madd(F3 a, float t, F3 d) { F3 r{a.x + t * d.x, a.y + t * d.y, a.z + t * d.z}; return r; }

__device__ __forceinline__ float sdiv(float x, float y) {
  return x / ((fabsf(y) < 1e-12f) ? 1e-12f : y);
}

// Ericson closest-point-on-triangle sq-distance, reference's override order.
__device__ float tri_sqdist(F3 p, F3 a, F3 b, F3 c) {
  F3 ab = f3sub(b, a), ac = f3sub(c, a), ap = f3sub(p, a);
  float d1 = f3dot(ab, ap), d2 = f3dot(ac, ap);
  F3 bp = f3sub(p, b);
  float d3 = f3dot(ab, bp), d4 = f3dot(ac, bp);
  F3 cp = f3sub(p, c);
  float d5 = f3dot(ab, cp), d6 = f3dot(ac, cp);
  float va = d3 * d6 - d5 * d4;
  float vb = d5 * d2 - d1 * d6;
  float vc = d1 * d4 - d3 * d2;
  F3 p_ab = f3madd(a, sdiv(d1, d1 - d3), ab);
  F3 p_ac = f3madd(a, sdiv(d2, d2 - d6), ac);
  F3 bc = f3sub(c, b);
  F3 p_bc = f3madd(b, sdiv(d4 - d3, (d4 - d3) + (d5 - d6)), bc);
  float inv = sdiv(1.0f, va + vb + vc);
  F3 p_face;
  p_face.x = a.x + vb * inv * ab.x + vc * inv * ac.x;
  p_face.y = a.y + vb * inv * ab.y + vc * inv * ac.y;
  p_face.z = a.z + vb * inv * ab.z + vc * inv * ac.z;
  F3 cpnt = p_face;
  if (va <= 0.0f && (d4 - d3) >= 0.0f && (d5 - d6) >= 0.0f) cpnt = p_bc;
  if (vb <= 0.0f && d2 >= 0.0f && d6 <= 0.0f) cpnt = p_ac;
  if (vc <= 0.0f && d1 >= 0.0f && d3 <= 0.0f) cpnt = p_ab;
  if (d6 >= 0.0f && d5 <= d6) cpnt = c;
  if (d3 >= 0.0f && d4 <= d3) cpnt = b;
  if (d1 <= 0.0f && d2 <= 0.0f) cpnt = a;
  F3 diff = f3sub(p, cpnt);
  return f3dot(diff, diff);
}

// ---------------------------------------------------------------------------
// Kernel 0: packed per-face record {cx,cy,cz,|c|^2, nx,ny,nz,0} + |vert|^2
// 32B records -> TDM-friendly contiguous 1D tensor, b128-loadable.
// ---------------------------------------------------------------------------
__global__ void mt_prep(const float* __restrict__ verts,
                        const int* __restrict__ faces,
                        const float* __restrict__ fnorm,
                        float* __restrict__ pk,
                        float* __restrict__ ppsq,
                        int V, int F) {
  int t = blockIdx.x * blockDim.x + threadIdx.x;
  if (t < F) {
    int i0 = faces[3 * t + 0], i1 = faces[3 * t + 1], i2 = faces[3 * t + 2];
    float cx = (verts[3 * i0 + 0] + verts[3 * i1 + 0] + verts[3 * i2 + 0]) / 3.0f;
    float cy = (verts[3 * i0 + 1] + verts[3 * i1 + 1] + verts[3 * i2 + 1]) / 3.0f;
    float cz = (verts[3 * i0 + 2] + verts[3 * i1 + 2] + verts[3 * i2 + 2]) / 3.0f;
    float* r = pk + 8 * (size_t)t;
    r[0] = cx; r[1] = cy; r[2] = cz;
    r[3] = cx * cx + cy * cy + cz * cz;
    r[4] = fnorm[3 * t + 0]; r[5] = fnorm[3 * t + 1]; r[6] = fnorm[3 * t + 2];
    r[7] = 0.0f;
  }
  if (t < V) {
    float x = verts[3 * t + 0], y = verts[3 * t + 1], z = verts[3 * t + 2];
    ppsq[t] = x * x + y * y + z * z;
  }
}

// ---------------------------------------------------------------------------
// Kernel 1: per-vertex distance stats -> scale = R/std (ddof=1)
// One wave per 16-vertex strip; WMMA f32 16x16x4 computes the p.c tile.
// ---------------------------------------------------------------------------
__global__ void mt_stats(const float* __restrict__ verts,
                         const float* __restrict__ pk,
                         const float* __restrict__ ppsq,
                         float* __restrict__ scale_out,
                         int V, int F) {
  const int lane = threadIdx.x;
  const int half = lane >> 4;
  const int nlane = lane & 15;
  const int v0 = blockIdx.x * 16;
  if (v0 + 16 > V) return;

  int mrow = v0 + nlane;
  v2f aP;
  if (half == 0) { aP.x = verts[3 * mrow + 0]; aP.y = verts[3 * mrow + 1]; }
  else           { aP.x = verts[3 * mrow + 2]; aP.y = 0.0f; }

  float pp8[8];
#pragma unroll
  for (int j = 0; j < 8; ++j) pp8[j] = ppsq[v0 + 8 * half + j];

  float s1[8], s2[8];
#pragma unroll
  for (int j = 0; j < 8; ++j) { s1[j] = 0.0f; s2[j] = 0.0f; }

  for (int f0 = 0; f0 < F; f0 += 16) {
    int f = f0 + nlane;
    __builtin_prefetch(pk + 8 * (size_t)(f + 16), 0, 3);    // global_prefetch_b8
    float4 r0 = *(const float4*)(pk + 8 * (size_t)f);       // global_load_b128
    v2f bC;
    if (half == 0) { bC.x = r0.x; bC.y = r0.y; }
    else           { bC.x = r0.z; bC.y = 0.0f; }
    float cc = r0.w;

    v8f acc = {};
    acc = wmma_f32_16x16x4(aP, bC, acc);
#pragma unroll
    for (int j = 0; j < 8; ++j) {
      float sq = fmaxf(pp8[j] + cc - 2.0f * acc[j], 0.0f);
      float ec = __builtin_amdgcn_sqrtf(sq);   // raw v_sqrt_f32 (ranking/stats)
      s1[j] += ec;
      s2[j] += ec * ec;
    }
  }

#pragma unroll
  for (int j = 0; j < 8; ++j) {
#pragma unroll
    for (int m = 1; m <= 8; m <<= 1) {
      s1[j] += __shfl_xor(s1[j], m, 32);
      s2[j] += __shfl_xor(s2[j], m, 32);
    }
  }
  if (nlane == 0) {
    float fn = (float)F;
#pragma unroll
    for (int j = 0; j < 8; ++j) {
      float var = (s2[j] - s1[j] * s1[j] / fn) / (fn - 1.0f);
      scale_out[v0 + 8 * half + j] = RCOEF / __builtin_amdgcn_sqrtf(fmaxf(var, 1e-30f));
    }
  }
}

// ---------------------------------------------------------------------------
// Kernel 2: TDM double-buffered face blocks -> 2x WMMA D-tiles -> per-row
//           streaming top-10-smallest. One wave per 16-vertex strip.
// ---------------------------------------------------------------------------
__global__ void mt_topk(const float* __restrict__ verts,
                        const float* __restrict__ vnorm,
                        const float* __restrict__ pk,
                        const float* __restrict__ ppsq,
                        const float* __restrict__ scale_in,
                        int* __restrict__ topk,
                        int V, int F) {
  __shared__ float stage[2][TILE_F * 8];   // 2 x 4KB TDM staging
  __shared__ float dtile[16][17];
  const int lane = threadIdx.x;
  const int half = lane >> 4;
  const int nlane = lane & 15;
  const int v0 = blockIdx.x * 16;
  if (v0 + 16 > V) return;

  int mrow = v0 + nlane;
  v2f aP, aN;
  if (half == 0) {
    aP.x = verts[3 * mrow + 0];  aP.y = verts[3 * mrow + 1];
    aN.x = -vnorm[3 * mrow + 0]; aN.y = -vnorm[3 * mrow + 1];
  } else {
    aP.x = verts[3 * mrow + 2];  aP.y = 0.0f;
    aN.x = -vnorm[3 * mrow + 2]; aN.y = 0.0f;
  }

  float pp8[8], sc8[8];
#pragma unroll
  for (int j = 0; j < 8; ++j) {
    pp8[j] = ppsq[v0 + 8 * half + j];
    sc8[j] = scale_in[v0 + 8 * half + j];
  }

  float bd[KSEL]; int bi[KSEL];
#pragma unroll
  for (int q = 0; q < KSEL; ++q) { bd[q] = 3.0e38f; bi[q] = 0; }

  const unsigned total_elems = (unsigned)F * 8u;
  const unsigned lds0 = (unsigned)(unsigned long long)(uintptr_t)&stage[0][0];
  const unsigned lds1 = (unsigned)(unsigned long long)(uintptr_t)&stage[1][0];
  const int ntile = F / TILE_F;

  if (ntile > 0)
    tdm_load_1d((unsigned long long)(uintptr_t)pk, lds0, TILE_F * 8, total_elems);

  int cur = 0;
  for (int t = 0; t < ntile; ++t) {
    if (t + 1 < ntile) {
      unsigned long long ga =
          (unsigned long long)(uintptr_t)(pk + (size_t)(t + 1) * TILE_F * 8);
      tdm_load_1d(ga, cur ? lds0 : lds1, TILE_F * 8, total_elems);
      __builtin_amdgcn_s_wait_tensorcnt(1);   // block t landed in LDS
    } else {
      __builtin_amdgcn_s_wait_tensorcnt(0);
    }
    __syncthreads();

    for (int s = 0; s < TILE_F / 16; ++s) {
      const float* rec = &stage[cur][(unsigned)(s * 16 + nlane) * 8u];
      float4 r0 = *(const float4*)rec;         // ds_load_b128: cent + |c|^2
      float4 r1 = *(const float4*)(rec + 4);   // ds_load_b128: face normal
      v2f bC, bN;
      if (half == 0) { bC.x = r0.x; bC.y = r0.y; bN.x = r1.x; bN.y = r1.y; }
      else           { bC.x = r0.z; bC.y = 0.0f; bN.x = r1.z; bN.y = 0.0f; }
      float cc = r0.w;

      v8f accP = {}; accP = wmma_f32_16x16x4(aP, bC, accP);  // p . centroid
      v8f accN = {}; accN = wmma_f32_16x16x4(aN, bN, accN);  // n_inv . normal

#pragma unroll
      for (int j = 0; j < 8; ++j) {
        float sq = fmaxf(pp8[j] + cc - 2.0f * accP[j], 0.0f);
        float ec = __builtin_amdgcn_sqrtf(sq);
        // per-row mean shift of ec cannot change the per-row ranking
        float D = (1.0f - accN[j]) + sc8[j] * ec;
        dtile[8 * half + j][nlane] = D;
      }
      __syncthreads();

      if (lane < 16) {
        int fb = t * TILE_F + s * 16;
#pragma unroll
        for (int i = 0; i < 16; ++i) {
          float d = dtile[lane][i];
          int fi = fb + i;
          if (d < bd[KSEL - 1]) {
            int pos = KSEL - 1;
#pragma unroll
            for (int q = KSEL - 2; q >= 0; --q)
              if (d < bd[q]) pos = q;
#pragma unroll
            for (int q = KSEL - 1; q >= 1; --q)
              if (q > pos) { bd[q] = bd[q - 1]; bi[q] = bi[q - 1]; }
            bd[pos] = d; bi[pos] = fi;   // stable: ties keep earlier index
          }
        }
      }
      __syncthreads();
    }
    cur ^= 1;
  }

  // remainder faces (F not a multiple of TILE_F): direct global path
  for (int f0 = ntile * TILE_F; f0 < F; f0 += 16) {
    int f = f0 + nlane;
    float4 r0 = *(const float4*)(pk + 8 * (size_t)f);
    float4 r1 = *(const float4*)(pk + 8 * (size_t)f + 4);
    v2f bC, bN;
    if (half == 0) { bC.x = r0.x; bC.y = r0.y; bN.x = r1.x; bN.y = r1.y; }
    else           { bC.x = r0.z; bC.y = 0.0f; bN.x = r1.z; bN.y = 0.0f; }
    float cc = r0.w;
    v8f accP = {}; accP = wmma_f32_16x16x4(aP, bC, accP);
    v8f accN = {}; accN = wmma_f32_16x16x4(aN, bN, accN);
#pragma unroll
    for (int j = 0; j < 8; ++j) {
      float sq = fmaxf(pp8[j] + cc - 2.0f * accP[j], 0.0f);
      float D = (1.0f - accN[j]) + sc8[j] * __builtin_amdgcn_sqrtf(sq);
      dtile[8 * half + j][nlane] = D;
    }
    __syncthreads();
    if (lane < 16) {
#pragma unroll
      for (int i = 0; i < 16; ++i) {
        float d = dtile[lane][i];
        int fi = f0 + i;
        if (d < bd[KSEL - 1]) {
          int pos = KSEL - 1;
#pragma unroll
          for (int q = KSEL - 2; q >= 0; --q)
            if (d < bd[q]) pos = q;
#pragma unroll
          for (int q = KSEL - 1; q >= 1; --q)
            if (q > pos) { bd[q] = bd[q - 1]; bi[q] = bi[q - 1]; }
          bd[pos] = d; bi[pos] = fi;
        }
      }
    }
    __syncthreads();
  }

  if (lane < 16) {
#pragma unroll
    for (int q = 0; q < KSEL; ++q) topk[(v0 + lane) * KSEL + q] = bi[q];
  }
}

// ---------------------------------------------------------------------------
// Kernel 3: iv re-rank (stable desc bubble sort == top_k(iv,10)), Ericson
//           distance, first-occurrence argmin, outputs.
// ---------------------------------------------------------------------------
__global__ void mt_final(const float* __restrict__ verts,
                         const float* __restrict__ vnorm,
                         const int* __restrict__ faces,
                         const float* __restrict__ fnorm,
                         const int* __restrict__ topk,
                         float* __restrict__ out,
                         int V) {
  int v = blockIdx.x * blockDim.x + threadIdx.x;
  if (v >= V) return;

  F3 p{verts[3 * v + 0], verts[3 * v + 1], verts[3 * v + 2]};
  F3 ninv{-vnorm[3 * v + 0], -vnorm[3 * v + 1], -vnorm[3 * v + 2]};

  int fidx[KSEL]; float iv[KSEL];
#pragma unroll
  for (int k = 0; k < KSEL; ++k) {
    int f = topk[v * KSEL + k];
    fidx[k] = f;
    float t = ninv.x * fnorm[3 * f + 0] + ninv.y * fnorm[3 * f + 1] + ninv.z * fnorm[3 * f + 2];
    iv[k] = (t > 0.5f) ? t : -1.0f;
  }
#pragma unroll
  for (int i = 0; i < KSEL; ++i) {
#pragma unroll
    for (int q = 0; q < KSEL - 1; ++q) {
      if (iv[q] < iv[q + 1]) {
        float tv = iv[q]; iv[q] = iv[q + 1]; iv[q + 1] = tv;
        int tf = fidx[q]; fidx[q] = fidx[q + 1]; fidx[q + 1] = tf;
      }
    }
  }

  float bestS = 3.0e38f; int bestK = 0;
#pragma unroll
  for (int k = 0; k < KSEL; ++k) {
    int f = fidx[k];
    int i0 = faces[3 * f + 0], i1 = faces[3 * f + 1], i2 = faces[3 * f + 2];
    F3 a{verts[3 * i0 + 0], verts[3 * i0 + 1], verts[3 * i0 + 2]};
    F3 b{verts[3 * i1 + 0], verts[3 * i1 + 1], verts[3 * i1 + 2]};
    F3 c{verts[3 * i2 + 0], verts[3 * i2 + 1], verts[3 * i2 + 2]};
    float s = tri_sqdist(p, a, b, c);
    if (s < bestS) { bestS = s; bestK = k; }
  }

  int f = fidx[bestK];
  int i0 = faces[3 * f + 0], i1 = faces[3 * f + 1], i2 = faces[3 * f + 2];
  F3 cc;
  cc.x = (verts[3 * i0 + 0] + verts[3 * i1 + 0] + verts[3 * i2 + 0]) / 3.0f;
  cc.y = (verts[3 * i0 + 1] + verts[3 * i1 + 1] + verts[3 * i2 + 1]) / 3.0f;
  cc.z = (verts[3 * i0 + 2] + verts[3 * i1 + 2] + verts[3 * i2 + 2]) / 3.0f;
  F3 cn{fnorm[3 * f + 0], fnorm[3 * f + 1], fnorm[3 * f + 2]};
  F3 dv = f3sub(p, cc);

  out[0 * V + v] = bestS;                 // squared distance
  out[1 * V + v] = sqrtf(f3dot(dv, dv));  // |p - centroid| (precise)
  out[2 * V + v] = (float)f;              // closest face index
  out[3 * V + v] = -f3dot(dv, cn);        // sign
}

// ---------------------------------------------------------------------------
extern "C" void kernel_launch(void* const* d_in, const int* in_sizes, int n_in,
                              void* d_out, int out_size, void* d_ws, size_t ws_size,
                              hipStream_t stream) {
  const float* verts = (const float*)d_in[0];
  const float* vnorm = (const float*)d_in[1];
  const int*   faces = (const int*)d_in[2];
  const float* fnorm = (const float*)d_in[3];
  const int V = in_sizes[0] / 3;
  const int F = in_sizes[2] / 3;

  float* pk    = (float*)d_ws;            // 8F packed records
  float* ppsq  = pk + (size_t)8 * F;      // V
  float* scale = ppsq + V;                // V
  int*   topk  = (int*)(scale + V);       // KSEL*V

  float* out = (float*)d_out;

  int np = (V > F) ? V : F;
  mt_prep<<<(np + 255) / 256, 256, 0, stream>>>(verts, faces, fnorm, pk, ppsq, V, F);
  mt_stats<<<V / 16, 32, 0, stream>>>(verts, pk, ppsq, scale, V, F);
  mt_topk<<<V / 16, 32, 0, stream>>>(verts, vnorm, pk, ppsq, scale, topk, V, F);
  mt_final<<<(V + 63) / 64, 64, 0, stream>>>(verts, vnorm, faces, fnorm, topk, out, V);
}